// BayesianLinear_54657753809053
// MI455X (gfx1250) — compile-verified
//
#include <hip/hip_runtime.h>
#include <math.h>

// Problem constants (match reference)
#define B_DIM   2048
#define OUT_DIM 256
#define IN_DIM  256

typedef float v2f __attribute__((ext_vector_type(2)));
typedef float v4f __attribute__((ext_vector_type(4)));
typedef float v8f __attribute__((ext_vector_type(8)));

// ---------------------------------------------------------------------------
// Kernel 0: precompute std = exp(0.5 * logvar) for weights (65536) and bias
// (256) into workspace. Avoids 134M transcendental ops in the hot stream.
// ---------------------------------------------------------------------------
__global__ void bl_std_precompute(const float* __restrict__ wlv,
                                  const float* __restrict__ blv,
                                  float* __restrict__ std_w,
                                  float* __restrict__ std_b) {
    int i = blockIdx.x * 256 + threadIdx.x;
    if (i < OUT_DIM * IN_DIM) std_w[i] = __expf(0.5f * wlv[i]);
    if (i < OUT_DIM)          std_b[i] = __expf(0.5f * blv[i]);
}

// ---------------------------------------------------------------------------
// Kernel 1: out[b,o] = sum_i x[b,i]*mu_w[o,i] + mu_b[o] + std_b[o]*eps_b[b,o]
// via V_WMMA_F32_16X16X4_F32. One wave per 16x16 (b,o) tile, K-loop of 64.
// Tiles: (2048/16)*(256/16) = 2048 waves = 256 blocks * 8 waves.
// A[m][k] = x[bt*16+m, k];  B[k][n] = mu_w[ot*16+n, k]  (B = mu^T).
// f32 A layout (ISA 7.12.2): lane<16 holds row M=lane, VGPR0/1 = K0/K1;
//                            lane>=16 holds row M=lane-16, VGPR0/1 = K2/K3.
// ---------------------------------------------------------------------------
__global__ void bl_wmma_mu_bias(const float* __restrict__ x,
                                const float* __restrict__ wmu,
                                const float* __restrict__ bmu,
                                const float* __restrict__ std_b,
                                const float* __restrict__ eps_b,
                                float* __restrict__ out) {
    const int wave = (blockIdx.x * blockDim.x + threadIdx.x) >> 5; // 0..2047
    const int lane = threadIdx.x & 31;
    const int bt   = wave / (OUT_DIM / 16);  // 0..127
    const int ot   = wave % (OUT_DIM / 16);  // 0..15
    const int half = lane >> 4;              // 0: K=0,1  1: K=2,3
    const int r    = lane & 15;              // A row (M) / B col (N)

    const float* xrow = x   + (size_t)(bt * 16 + r) * IN_DIM;
    const float* wrow = wmu + (size_t)(ot * 16 + r) * IN_DIM;

    v8f c = {};
#pragma unroll 8
    for (int kk = 0; kk < IN_DIM; kk += 4) {
        // 8-byte aligned v2f loads (kk%4==0, +2*half keeps even offset)
        v2f a = *(const v2f*)(xrow + kk + 2 * half);
        v2f b = *(const v2f*)(wrow + kk + 2 * half);
        // (neg_a, A, neg_b, B, c_mod, C, reuse_a, reuse_b)
        c = __builtin_amdgcn_wmma_f32_16x16x4_f32(
                false, a, false, b, (short)0, c, false, false);
    }

    // C/D layout: VGPR j -> (M = j + 8*half, N = lane&15)
    const int o  = ot * 16 + r;
    const float bm = bmu[o];
    const float bs = std_b[o];
#pragma unroll
    for (int j = 0; j < 8; ++j) {
        const int brow = bt * 16 + j + 8 * half;
        const size_t idx = (size_t)brow * OUT_DIM + o;
        out[idx] = c[j] + bm + bs * eps_b[idx];
    }
}

// ---------------------------------------------------------------------------
// Kernel 2: out[b,o] += sum_i std_w[o,i] * eps_w[b,o,i] * x[b,i].
// One wave per (b,o) pair: 32 lanes * 8 f32 = the full 256-element row.
// eps_w (536 MB, single use) is loaded non-temporally with coalesced b128;
// std_w row (1 KB, reused 2048x) and x row (1 KB, reused 256x) hit L2/WGP$.
// Wave32 butterfly reduction, lane 0 accumulates into d_out (kernel 1 ran
// earlier on the same stream, so the RMW is ordered).
// ---------------------------------------------------------------------------
__global__ void bl_eps_stream(const float* __restrict__ eps_w,
                              const float* __restrict__ std_w,
                              const float* __restrict__ x,
                              float* __restrict__ out) {
    const int p    = (blockIdx.x * blockDim.x + threadIdx.x) >> 5; // (b,o) pair
    const int lane = threadIdx.x & 31;
    const int b    = p >> 8;
    const int o    = p & (OUT_DIM - 1);

    const v4f* e4 = (const v4f*)eps_w + (size_t)p * (IN_DIM / 4);
    const v4f* s4 = (const v4f*)std_w + (size_t)o * (IN_DIM / 4);
    const v4f* x4 = (const v4f*)x     + (size_t)b * (IN_DIM / 4);

    float acc = 0.0f;
#pragma unroll
    for (int chunk = 0; chunk < 2; ++chunk) {
        const int idx = chunk * 32 + lane;
        v4f e  = __builtin_nontemporal_load(e4 + idx); // TH=NT: one-shot stream
        v4f s  = s4[idx];
        v4f xv = x4[idx];
        acc = fmaf(e.x, s.x * xv.x, acc);
        acc = fmaf(e.y, s.y * xv.y, acc);
        acc = fmaf(e.z, s.z * xv.z, acc);
        acc = fmaf(e.w, s.w * xv.w, acc);
    }

    // wave32 reduction
#pragma unroll
    for (int m = 16; m >= 1; m >>= 1)
        acc += __shfl_xor(acc, m, 32);

    if (lane == 0)
        out[p] += acc;
}

// ---------------------------------------------------------------------------
extern "C" void kernel_launch(void* const* d_in, const int* in_sizes, int n_in,
                              void* d_out, int out_size, void* d_ws, size_t ws_size,
                              hipStream_t stream) {
    const float* x     = (const float*)d_in[0];
    const float* wmu   = (const float*)d_in[1];
    const float* wlv   = (const float*)d_in[2];
    const float* bmu   = (const float*)d_in[3];
    const float* blv   = (const float*)d_in[4];
    const float* eps_w = (const float*)d_in[5];
    const float* eps_b = (const float*)d_in[6];
    float* out = (float*)d_out;

    // Workspace: std_w (256 KB) + std_b (1 KB)
    float* std_w = (float*)d_ws;
    float* std_b = std_w + OUT_DIM * IN_DIM;

    // 1) sigma = exp(0.5*logvar)
    bl_std_precompute<<<(OUT_DIM * IN_DIM) / 256, 256, 0, stream>>>(
        wlv, blv, std_w, std_b);

    // 2) WMMA GEMM x @ mu^T + bias terms -> overwrites d_out (deterministic)
    //    2048 tiles / 8 waves-per-block = 256 blocks
    bl_wmma_mu_bias<<<(B_DIM / 16) * (OUT_DIM / 16) / 8, 256, 0, stream>>>(
        x, wmu, bmu, std_b, eps_b, out);

    // 3) stream eps_w: one wave per (b,o) pair -> 524288 waves / 8 = 65536 blocks
    bl_eps_stream<<<(B_DIM * OUT_DIM) / 8, 256, 0, stream>>>(
        eps_w, std_w, x, out);
}